// Conv3_5738076307876
// MI455X (gfx1250) — compile-verified
//
#include <hip/hip_runtime.h>

typedef __attribute__((ext_vector_type(2))) float v2f;
typedef __attribute__((ext_vector_type(8))) float v8f;

#define C_CH 10    // channels (in = out = depth)
#define HW   128
#define KSZ  5
#define KPAD 128   // 125 taps (i',kh,kw) padded to 128 (multiple of WMMA K=4)
#define NPAD 16    // 10 output channels padded to WMMA M=16
#define XROWS 26   // 25 (i',kh) rows + 1 zero pad row for taps k=125..127
#define XCOLS 132  // 128 + 2-halo each side

// tap k = i'*25 + kh*5 + kw  ->  LDS row/col base.  For k in [125,128) this
// decodes to row 25 (the zeroed pad row), so one formula covers all of KPAD.
__host__ __device__ constexpr int tapbase(int k) {
    int ip  = k / 25;
    int rem = k - ip * 25;
    int kh  = rem / 5;
    int kw  = rem - kh * 5;
    return (ip * 5 + kh) * XCOLS + kw;
}

// ---------------------------------------------------------------------------
// Pass 1: pack weights into Wpack[d][k][o], k = i'*25 + kh*5 + kw, i = d-2+i'.
// Zero for clipped i, for o >= 10, and for k >= 125.  10*128*16 f32 = 80 KB.
// ---------------------------------------------------------------------------
__global__ void pack_weights(const float* __restrict__ W, float* __restrict__ Wpack) {
    int idx = blockIdx.x * blockDim.x + threadIdx.x;
    if (idx >= C_CH * KPAD * NPAD) return;
    int o = idx & 15;
    int k = (idx >> 4) & 127;
    int d = idx >> 11;
    float v = 0.f;
    if (o < C_CH && k < 125) {
        int ip  = k / 25;            // i' == kd
        int rem = k - ip * 25;
        int kh  = rem / 5;
        int kw  = rem - kh * 5;
        int i   = d - 2 + ip;
        if (i >= 0 && i < C_CH) {
            // W layout: [o][i][kd][kh][kw]
            v = W[(((o * C_CH + i) * KSZ + ip) * KSZ + kh) * KSZ + kw];
        }
    }
    Wpack[idx] = v;
}

// ---------------------------------------------------------------------------
// Pass 2: one workgroup per (b, d, h).  Computes out[b, 0..9, d, h, 0..127]
// as a 16(M=o) x 128(N=w) x 128(K=taps) GEMM with V_WMMA_F32_16X16X4_F32.
// 4 waves x 2 N-tiles each; fully unrolled K loop -> 64 WMMA per wave,
// with all tap decodes folded to constants (one v_cndmask per operand).
// ---------------------------------------------------------------------------
__global__ __launch_bounds__(128) void conv_diag3d(
    const float* __restrict__ x,      // [B][10][128][128]
    const float* __restrict__ Wpack,  // [10][128][16]
    const float* __restrict__ bias,   // [10]
    float* __restrict__ out)          // [B][10][10][128][128]
{
    __shared__ float xs[XROWS * XCOLS];   // x rows with halo; col index = w + 2
    __shared__ float wl[KPAD * NPAD];     // [k][o]

    const int h   = blockIdx.x;
    const int d   = blockIdx.y;
    const int b   = blockIdx.z;
    const int tid = threadIdx.x;

    // Stage this d's weight slab (8 KB).
    const float* wp = Wpack + d * (KPAD * NPAD);
    for (int e = tid; e < KPAD * NPAD; e += 128) wl[e] = wp[e];

    // Stage x rows (13.7 KB), zero-filled outside the valid tensor.
    // Incremental row/col tracking (no division): stride 128 < XCOLS wraps <= once.
    {
        int row = 0, col = tid;
        const float* xb = x + (size_t)b * C_CH * HW * HW;
        for (int e = tid; e < XROWS * XCOLS; e += 128) {
            int ip = row / 5;            // i' (row < 26 -> tiny constant-folded div)
            int kh = row - ip * 5;
            int i  = d - 2 + ip;
            int hy = h - 2 + kh;
            int w  = col - 2;
            float v = 0.f;
            if (row < 25 && i >= 0 && i < C_CH && hy >= 0 && hy < HW && w >= 0 && w < HW)
                v = xb[(i * HW + hy) * HW + w];
            xs[e] = v;
            col += 128;
            if (col >= XCOLS) { col -= XCOLS; row += 1; }
        }
    }
    __syncthreads();

    const int lane   = tid & 31;
    const int wave   = tid >> 5;
    const int lo     = lane & 15;        // N column (w) / M row (o) selector
    const bool hiLn  = (lane >= 16);     // lanes 16-31 hold K pair {k0+2, k0+3}

    // Loop-invariant LDS bases.
    const float* wlp = &wl[(hiLn ? 32 : 0) + lo];     // + k0*16 (constant) per step
    const float* xsp = &xs[(wave * 2) * 16 + lo];     // + base (cndmask) per step; tile1 = +16

    v8f c0 = {};
    v8f c1 = {};

#pragma unroll
    for (int k0 = 0; k0 < KPAD; k0 += 4) {
        // A fragment (16x4, M = o): lane holds row o = lo; regs K = kA, kA+1.
        v2f a;
        a.x = wlp[k0 * NPAD];            // constant LDS byte offsets
        a.y = wlp[k0 * NPAD + NPAD];

        // B fragment (4x16, N = w): lane holds col w; regs K = kA, kA+1.
        // tapbase(k0..k0+3) are compile-time constants -> one select per operand.
        const int base0 = hiLn ? tapbase(k0 + 2) : tapbase(k0 + 0);
        const int base1 = hiLn ? tapbase(k0 + 3) : tapbase(k0 + 1);

        v2f b0, b1;
        b0.x = xsp[base0];               // tile 0
        b0.y = xsp[base1];
        b1.x = xsp[base0 + 16];          // tile 1: constant +64 B offset
        b1.y = xsp[base1 + 16];

        // D = A x B + C   (8 args: neg_a, A, neg_b, B, c_mod, C, reuse_a, reuse_b)
        c0 = __builtin_amdgcn_wmma_f32_16x16x4_f32(false, a, false, b0, (short)0, c0, false, false);
        c1 = __builtin_amdgcn_wmma_f32_16x16x4_f32(false, a, false, b1, (short)0, c1, false, false);
    }

    // D layout: reg r, lanes 0-15 -> M = r, lanes 16-31 -> M = r + 8; N = lane & 15.
    const int hiM = hiLn ? 8 : 0;
    const int wb0 = (wave * 2) * 16;
#pragma unroll
    for (int r = 0; r < 8; ++r) {
        int o = r + hiM;
        if (o < C_CH) {
            float bo = bias[o];
            float* dst = out + (((size_t)(b * C_CH + o) * C_CH + d) * HW + h) * HW + wb0 + lo;
            dst[0]  = c0[r] + bo;
            dst[16] = c1[r] + bo;
        }
    }
}

// ---------------------------------------------------------------------------
extern "C" void kernel_launch(void* const* d_in, const int* in_sizes, int n_in,
                              void* d_out, int out_size, void* d_ws, size_t ws_size,
                              hipStream_t stream) {
    const float* x    = (const float*)d_in[0];   // [B,10,128,128]
    const float* W    = (const float*)d_in[1];   // [10,10,5,5,5]
    const float* bias = (const float*)d_in[2];   // [10]
    float* out = (float*)d_out;

    int B = in_sizes[0] / (C_CH * HW * HW);

    float* Wpack = (float*)d_ws;                 // 80 KB scratch
    int nw = C_CH * KPAD * NPAD;
    pack_weights<<<(nw + 255) / 256, 256, 0, stream>>>(W, Wpack);

    dim3 grid(HW, C_CH, B);                      // (h, d, b)
    conv_diag3d<<<grid, 128, 0, stream>>>(x, Wpack, bias, out);
}